// TGNN_31877247271269
// MI455X (gfx1250) — compile-verified
//
#include <hip/hip_runtime.h>

typedef __attribute__((ext_vector_type(2))) float v2f;
typedef __attribute__((ext_vector_type(8))) float v8f;

#define HID   128
#define DIN   256
#define LROW  260   // 16x256 tile row stride in LDS (pad 4 -> stride%64==4, conflict-free)

// ---------------------------------------------------------------------------
// Kernel 1: zero the node accumulator h[N, 256]
// ---------------------------------------------------------------------------
__global__ __launch_bounds__(256) void tgnn_zero_kernel(float4* __restrict__ p, long long n4) {
    long long i = (long long)blockIdx.x * blockDim.x + threadIdx.x;
    if (i < n4) p[i] = make_float4(0.f, 0.f, 0.f, 0.f);
}

// ---------------------------------------------------------------------------
// Kernel 2: per-edge feature build + scatter-add into destination nodes.
// One block per edge; thread j owns feature j (j<128: raw edge feat,
// j>=128: cosine time encoding).
// ---------------------------------------------------------------------------
__global__ __launch_bounds__(256) void tgnn_edge_kernel(
    const int*   __restrict__ src,
    const int*   __restrict__ dst,
    const float* __restrict__ ef,
    const float* __restrict__ bt,
    const float* __restrict__ node_ts,
    const float* __restrict__ time_w,
    const float* __restrict__ time_b,
    float*       __restrict__ h)
{
    int e = blockIdx.x;
    int j = threadIdx.x;
    int s = src[e];
    int d = dst[e];
    float td = bt[e] - node_ts[s];
    float v;
    if (j < HID) {
        v = ef[(size_t)e * HID + j];
    } else {
        int k = j - HID;
        v = cosf(td * time_w[k] + time_b[k]);
    }
    atomicAdd(&h[(size_t)d * DIN + j], v);
}

// ---------------------------------------------------------------------------
// Kernel 3: fused predictor. Block = one 16-row query tile, 8 waves.
// Wave w computes columns [16w, 16w+16) of h_src/h_pos/h_neg via
// V_WMMA_F32_16X16X4_F32 over K=256, then ReLU + W_out projection with a
// deterministic fixed-order reduction.
// ---------------------------------------------------------------------------
__global__ __launch_bounds__(256) void tgnn_predict_kernel(
    const float* __restrict__ h,
    const int*   __restrict__ s_idx,
    const int*   __restrict__ p_idx,
    const int*   __restrict__ n_idx,
    const float* __restrict__ W_src,
    const float* __restrict__ b_src,
    const float* __restrict__ W_dst,
    const float* __restrict__ b_dst,
    const float* __restrict__ W_out,
    const float* __restrict__ b_out,
    float*       __restrict__ out,
    int Q)
{
    __shared__ float sl[16 * LROW];
    __shared__ float pl[16 * LROW];
    __shared__ float nl[16 * LROW];
    __shared__ float red_pos[8 * 256];   // [wave][r][lane]
    __shared__ float red_neg[8 * 256];

    const int tile = blockIdx.x;
    const int t    = threadIdx.x;

    // Gather the three 16x256 embedding tiles into LDS.
    for (int i = t; i < 16 * DIN; i += 256) {
        int row = i >> 8;
        int col = i & 255;
        int qi  = tile * 16 + row;
        sl[row * LROW + col] = h[(size_t)s_idx[qi] * DIN + col];
        pl[row * LROW + col] = h[(size_t)p_idx[qi] * DIN + col];
        nl[row * LROW + col] = h[(size_t)n_idx[qi] * DIN + col];
    }
    __syncthreads();

    const int wave = t >> 5;
    const int lane = t & 31;
    const int half = lane >> 4;      // lane half selects K sub-pairs
    const int lm   = lane & 15;      // M for A, N for B/C/D
    const int n0   = wave * 16;      // this wave's output-column base

    v8f hs = {}, hp = {}, hn = {};

    const float* As = &sl[lm * LROW];
    const float* Ap = &pl[lm * LROW];
    const float* An = &nl[lm * LROW];

    for (int k0 = 0; k0 < DIN; k0 += 4) {
        int ka = k0 + 2 * half;
        // A 16x4 f32 layout: VGPR0 = K(ka), VGPR1 = K(ka+1), lane-half picks {0,1} vs {2,3}
        v2f a_s; a_s.x = As[ka]; a_s.y = As[ka + 1];
        v2f a_p; a_p.x = Ap[ka]; a_p.y = Ap[ka + 1];
        v2f a_n; a_n.x = An[ka]; a_n.y = An[ka + 1];
        // B 4x16 f32: row K striped across lanes (N = lm), same K split as A
        v2f b_s; b_s.x = W_src[ka * HID + n0 + lm]; b_s.y = W_src[(ka + 1) * HID + n0 + lm];
        v2f b_d; b_d.x = W_dst[ka * HID + n0 + lm]; b_d.y = W_dst[(ka + 1) * HID + n0 + lm];

        hs = __builtin_amdgcn_wmma_f32_16x16x4_f32(false, a_s, false, b_s, (short)0, hs, false, false);
        hp = __builtin_amdgcn_wmma_f32_16x16x4_f32(false, a_p, false, b_d, (short)0, hp, false, false);
        hn = __builtin_amdgcn_wmma_f32_16x16x4_f32(false, a_n, false, b_d, (short)0, hn, false, false);
    }

    // C/D layout: VGPR r, lane -> element (M = r + 8*half, N = n0 + lm).
    const float bias = b_src[n0 + lm] + b_dst[n0 + lm];
    const float wo   = W_out[n0 + lm];

    #pragma unroll
    for (int r = 0; r < 8; ++r) {
        float pv = fmaxf(hs[r] + hp[r] + bias, 0.f) * wo;
        float nv = fmaxf(hs[r] + hn[r] + bias, 0.f) * wo;
        red_pos[wave * 256 + r * 32 + lane] = pv;
        red_neg[wave * 256 + r * 32 + lane] = nv;
    }
    __syncthreads();

    // Deterministic fixed-order reduction: thread m sums its row's 128 terms.
    if (t < 16) {
        int mh = t >> 3;          // which lane-half produced row t
        int r  = t & 7;           // accumulator register index
        float accp = b_out[0];
        float accn = b_out[0];
        for (int w = 0; w < 8; ++w) {
            #pragma unroll
            for (int l = 0; l < 16; ++l) {
                int idx = w * 256 + r * 32 + mh * 16 + l;
                accp += red_pos[idx];
                accn += red_neg[idx];
            }
        }
        out[tile * 16 + t]     = accp;   // pos logits
        out[Q + tile * 16 + t] = accn;   // neg logits
    }
}

// ---------------------------------------------------------------------------
extern "C" void kernel_launch(void* const* d_in, const int* in_sizes, int n_in,
                              void* d_out, int out_size, void* d_ws, size_t ws_size,
                              hipStream_t stream) {
    const int*   src     = (const int*)  d_in[0];
    const int*   dst     = (const int*)  d_in[1];
    const float* ef      = (const float*)d_in[2];
    const float* bt      = (const float*)d_in[3];
    const float* node_ts = (const float*)d_in[4];
    const int*   s_idx   = (const int*)  d_in[5];
    const int*   p_idx   = (const int*)  d_in[6];
    const int*   n_idx   = (const int*)  d_in[7];
    const float* time_w  = (const float*)d_in[8];
    const float* time_b  = (const float*)d_in[9];
    const float* W_src   = (const float*)d_in[10];
    const float* b_src   = (const float*)d_in[11];
    const float* W_dst   = (const float*)d_in[12];
    const float* b_dst   = (const float*)d_in[13];
    const float* W_out   = (const float*)d_in[14];
    const float* b_out   = (const float*)d_in[15];

    float* out = (float*)d_out;
    float* h   = (float*)d_ws;             // [N_NODES, 256] accumulator

    const int NN = in_sizes[4];            // node_ts -> N_NODES
    const int NE = in_sizes[0];            // src -> N_EDGES
    const int Q  = in_sizes[5];            // s_idx -> Q

    // 1) zero node accumulator
    long long n4 = (long long)NN * DIN / 4;
    int zgrid = (int)((n4 + 255) / 256);
    tgnn_zero_kernel<<<zgrid, 256, 0, stream>>>((float4*)h, n4);

    // 2) edge feature build + scatter-add
    tgnn_edge_kernel<<<NE, 256, 0, stream>>>(src, dst, ef, bt, node_ts, time_w, time_b, h);

    // 3) fused WMMA predictor
    tgnn_predict_kernel<<<Q / 16, 256, 0, stream>>>(h, s_idx, p_idx, n_idx,
                                                    W_src, b_src, W_dst, b_dst,
                                                    W_out, b_out, out, Q);
}